// ScaledDotProductAttention_24515673325677
// MI455X (gfx1250) — compile-verified
//
#include <hip/hip_runtime.h>

// ScaledDotProductAttention for B=2,H=16,S=2048,D=128, fp32, TEMPERATURE=1.
// Outputs (concatenated): output [B,H,S,D] then attn [B,H,S,S], both fp32.
//
// Fused design, one 16-row q-tile per 256-thread workgroup (8 wave32):
//   Phase A: scores = Q K^T via v_wmma_f32_16x16x4_f32 (fp32 in/out, full precision)
//   Phase B: masked+scaled row softmax in LDS, attn streamed out with NT stores
//   Phase C: out = P V via v_wmma_f32_16x16x4_f32
// Roofline: ~640MB+ traffic vs 68.7 GFLOP -> memory bound; NT hints keep the
// single-use attn/mask streams from evicting the K/V tiles reused across the
// 128 q-tile workgroups per (b,h) in L2 (192MB).

typedef float v2f __attribute__((ext_vector_type(2)));
typedef float v4f __attribute__((ext_vector_type(4)));
typedef float v8f __attribute__((ext_vector_type(8)));

#define BATCH 2
#define HEADS 16
#define SEQ   2048
#define DIM   128
#define QT    16        // q rows per workgroup
#define NWAVE 8         // waves per workgroup (wave32)
#define KPW   (SEQ / NWAVE)   // 256 key columns per wave in phase A
#define LDQ   132       // padded row stride for Q tile (bank-conflict-free col reads)
#define LDP   2052      // padded row stride for score/prob tile
#define LDV   132       // padded row stride for K/V staging

// LDS partition (floats):
//   Qs    : QT*LDQ           = 2112
//   P     : QT*LDP           = 32832
//   Stage : max(8*16*LDV, 128*LDV) = 16896   (K slabs in phase A, V chunk in phase C,
//                                             softmax reduction scratch in phase B)
#define SMEM_FLOATS (QT*LDQ + QT*LDP + 16896)

__global__ __launch_bounds__(256)
void sdpa_f32_wmma_kernel(const float* __restrict__ Q,
                          const float* __restrict__ K,
                          const float* __restrict__ V,
                          const int*   __restrict__ mask,
                          float* __restrict__ Out,
                          float* __restrict__ Attn)
{
    extern __shared__ float smem[];
    float* Qs    = smem;                 // [QT][LDQ]
    float* P     = Qs + QT * LDQ;        // [QT][LDP]
    float* Stage = P  + QT * LDP;        // 16896 floats

    const int tid  = threadIdx.x;
    const int wave = tid >> 5;
    const int lane = tid & 31;
    const int l16  = lane & 15;
    const int half = lane >> 4;          // 0: lanes 0-15, 1: lanes 16-31

    const int wg    = blockIdx.x;        // 0 .. B*H*(S/QT)-1
    const int qtile = wg % (SEQ / QT);
    const int bh    = wg / (SEQ / QT);   // 0 .. 31
    const int qbase = qtile * QT;

    const float* Qp = Q + (size_t)bh * SEQ * DIM;
    const float* Kp = K + (size_t)bh * SEQ * DIM;
    const float* Vp = V + (size_t)bh * SEQ * DIM;
    float* Op = Out  + (size_t)bh * SEQ * DIM;
    float* Ap = Attn + (size_t)bh * SEQ * SEQ;

    // ---- Load 16x128 Q tile into LDS (coalesced 128-bit) ----
    for (int i = tid; i < QT * DIM / 4; i += 256) {
        const int r = (i * 4) / DIM, c = (i * 4) % DIM;
        const v4f q4 = *(const v4f*)(Qp + (size_t)(qbase + r) * DIM + c);
        float* dst = Qs + r * LDQ + c;
        dst[0] = q4[0]; dst[1] = q4[1]; dst[2] = q4[2]; dst[3] = q4[3];
    }
    __syncthreads();

    // ================= Phase A: scores = Q @ K^T =================
    // Each wave owns key columns [wave*KPW, wave*KPW+KPW), 16 tiles of 16.
    float* Ks = Stage + wave * (16 * LDV);   // per-wave 16x128 K slab
    for (int i = 0; i < KPW / 16; ++i) {
        const int nbase = wave * KPW + i * 16;
        // wave stages K[nbase..nbase+16) x 128 (2048 floats / 32 lanes)
        for (int t = lane; t < 16 * DIM / 4; t += 32) {
            const int r = (t * 4) / DIM, c = (t * 4) % DIM;
            const v4f k4 = *(const v4f*)(Kp + (size_t)(nbase + r) * DIM + c);
            float* dst = Ks + r * LDV + c;
            dst[0] = k4[0]; dst[1] = k4[1]; dst[2] = k4[2]; dst[3] = k4[3];
        }
        __syncthreads();

        v8f acc = {};
        for (int kd = 0; kd < DIM; kd += 4) {
            const int c0 = kd + half * 2;        // lanes 16-31 hold K=2,3
            v2f a, b;
            a[0] = Qs[l16 * LDQ + c0];  a[1] = Qs[l16 * LDQ + c0 + 1];  // A: M=l16, K=c0..c0+1
            b[0] = Ks[l16 * LDV + c0];  b[1] = Ks[l16 * LDV + c0 + 1];  // B: N=l16 (=key), K=c0..c0+1
            acc = __builtin_amdgcn_wmma_f32_16x16x4_f32(
                      false, a, false, b, (short)0, acc, false, false);
        }
        // C layout: VGPR v -> row (v + half*8), col l16
        #pragma unroll
        for (int v = 0; v < 8; ++v)
            P[(v + half * 8) * LDP + nbase + l16] = acc[v];
        __syncthreads();
    }

    // ================= Phase B: masked, scaled softmax per row =================
    {
        const int row = tid >> 4;    // 16 rows, 16 threads each
        const int t16 = tid & 15;
        float* red  = Stage;         // 16x16 reduction scratch (K slabs dead now)
        float* prow = P + row * LDP;
        const int* mrow = mask + (size_t)(qbase + row) * SEQ;
        const float scale = 0.08838834764831845f;  // 1/sqrt(128)

        float mx = -__builtin_inff();
        for (int k = t16; k < SEQ; k += 16) {
            float s = prow[k] * scale;
            const int m = __builtin_nontemporal_load(mrow + k);   // single-use stream
            s = (m == 0) ? -__builtin_inff() : s;
            prow[k] = s;
            mx = fmaxf(mx, s);
        }
        red[row * 16 + t16] = mx;
        __syncthreads();
        if (t16 == 0) {
            float m = red[row * 16];
            #pragma unroll
            for (int j = 1; j < 16; ++j) m = fmaxf(m, red[row * 16 + j]);
            red[row * 16] = m;
        }
        __syncthreads();
        mx = red[row * 16];
        __syncthreads();

        float sum = 0.f;
        for (int k = t16; k < SEQ; k += 16) {
            const float e = __expf(prow[k] - mx);
            prow[k] = e;
            sum += e;
        }
        red[row * 16 + t16] = sum;
        __syncthreads();
        if (t16 == 0) {
            float s = 0.f;
            #pragma unroll
            for (int j = 0; j < 16; ++j) s += red[row * 16 + j];
            red[row * 16] = 1.0f / s;
        }
        __syncthreads();
        const float rinv = red[row * 16];
        for (int k = t16; k < SEQ; k += 16) prow[k] *= rinv;
        __syncthreads();   // P finalized; red reads done (Stage reusable)
    }

    // ---- Write attn (coalesced 128-bit, non-temporal: never re-read) ----
    for (int i = tid; i < QT * SEQ / 4; i += 256) {
        const int r = (i * 4) / SEQ, c = (i * 4) % SEQ;
        const float* src = P + r * LDP + c;
        v4f p4; p4[0] = src[0]; p4[1] = src[1]; p4[2] = src[2]; p4[3] = src[3];
        __builtin_nontemporal_store(p4, (v4f*)(Ap + (size_t)(qbase + r) * SEQ + c));
    }

    // ================= Phase C: out = P @ V =================
    // Each wave computes the 16x16 output tile at columns [wave*16, wave*16+16).
    float* Vs = Stage;               // [128][LDV] shared V chunk
    v8f oacc = {};
    for (int kc = 0; kc < SEQ; kc += 128) {
        __syncthreads();             // prior chunk consumed before overwrite
        for (int i = tid; i < 128 * DIM / 4; i += 256) {
            const int r = (i * 4) / DIM, c = (i * 4) % DIM;
            const v4f v4 = *(const v4f*)(Vp + (size_t)(kc + r) * DIM + c);
            float* dst = Vs + r * LDV + c;
            dst[0] = v4[0]; dst[1] = v4[1]; dst[2] = v4[2]; dst[3] = v4[3];
        }
        __syncthreads();
        for (int kd = 0; kd < 128; kd += 4) {
            const int c0 = kd + half * 2;
            v2f a, b;
            a[0] = P[l16 * LDP + kc + c0];            // A: M=l16, K=kc+c0..+1
            a[1] = P[l16 * LDP + kc + c0 + 1];
            b[0] = Vs[c0 * LDV + wave * 16 + l16];    // B: K=c0, N=wave*16+l16
            b[1] = Vs[(c0 + 1) * LDV + wave * 16 + l16];
            oacc = __builtin_amdgcn_wmma_f32_16x16x4_f32(
                       false, a, false, b, (short)0, oacc, false, false);
        }
    }
    #pragma unroll
    for (int v = 0; v < 8; ++v)
        __builtin_nontemporal_store(oacc[v],
            Op + (size_t)(qbase + v + half * 8) * DIM + wave * 16 + l16);
}

extern "C" void kernel_launch(void* const* d_in, const int* in_sizes, int n_in,
                              void* d_out, int out_size, void* d_ws, size_t ws_size,
                              hipStream_t stream) {
    const float* Q    = (const float*)d_in[0];
    const float* K    = (const float*)d_in[1];
    const float* V    = (const float*)d_in[2];
    const int*   mask = (const int*)d_in[3];

    float* Out  = (float*)d_out;                              // [B,H,S,D]
    float* Attn = Out + (size_t)BATCH * HEADS * SEQ * DIM;    // [B,H,S,S]

    const int nblocks = BATCH * HEADS * (SEQ / QT);           // 4096
    const size_t smem = (size_t)SMEM_FLOATS * sizeof(float);  // 207360 B < 320KB/WGP
    sdpa_f32_wmma_kernel<<<nblocks, 256, smem, stream>>>(Q, K, V, mask, Out, Attn);
}